// GCN_23502061044172
// MI455X (gfx1250) — compile-verified
//
#include <hip/hip_runtime.h>
#include <hip/hip_bf16.h>
#include <math.h>

// Problem constants (match reference)
#define N_NODES  50000
#define N_EDGES  800000
#define F_IN     128
#define HID      64
#define N_CLS    8
#define N_GRAPHS 64

typedef __attribute__((ext_vector_type(2))) float v2f;
typedef __attribute__((ext_vector_type(8))) float v8f;

// ---------------------------------------------------------------------------
// utility: fill a float buffer with a constant
// ---------------------------------------------------------------------------
__global__ void gcn_fill(float* __restrict__ p, float v, int n) {
  int i = blockIdx.x * blockDim.x + threadIdx.x;
  if (i < n) p[i] = v;
}

// ---------------------------------------------------------------------------
// degree accumulation over real edges (deg pre-initialized to 1.0 for self-loop)
// ---------------------------------------------------------------------------
__global__ void gcn_deg_accum(const int* __restrict__ dst, float* __restrict__ deg, int e) {
  int i = blockIdx.x * blockDim.x + threadIdx.x;
  if (i < e) unsafeAtomicAdd(&deg[dst[i]], 1.0f);
}

__global__ void gcn_dinv_inplace(float* __restrict__ p, int n) {
  int i = blockIdx.x * blockDim.x + threadIdx.x;
  if (i < n) p[i] = 1.0f / sqrtf(p[i]);  // deg >= 1 always (self loops)
}

// ---------------------------------------------------------------------------
// WMMA f32 GEMM:  H[M,N] = X[M,K] @ W[K,N],  M = N_NODES (multiple of 16)
// K, N compile-time -> pure 32-bit shift addressing, unrolled wmma chain.
// W is staged once per block into LDS, transposed with +2 pad so each lane's
// {k, k+1} pair for its column is one aligned contiguous LDS read
// (lowers to ds_load_2addr_b64).
// VGPR layouts per CDNA5 ISA 7.12.2:
//   A 16x4:  lanes 0-15 -> M=lane, {v0,v1}={K0,K1}; lanes 16-31 -> {K2,K3}
//   B 4x16:  lanes 0-15 -> N=lane, {v0,v1}={K0,K1}; lanes 16-31 -> {K2,K3}
//   D 16x16: vgpr r: lanes 0-15 -> M=r, N=lane; lanes 16-31 -> M=r+8
// EXEC is all-1s at every WMMA: tile guard is wave-uniform (after the block-
// wide barrier). Full tiles (N%16==0) compile with zero masking; the ragged
// N=8 case uses clamped addresses + cndmask (no exec edits).
// ---------------------------------------------------------------------------
template <int K, int N>
__global__ void gcn_gemm_wmma(const float* __restrict__ X, const float* __restrict__ W,
                              float* __restrict__ H) {
  constexpr int  TILES_N = (N + 15) / 16;
  constexpr int  KP      = K + 2;              // row pad: keeps b64 align, spreads banks
  constexpr bool FULL    = (N % 16 == 0);
  __shared__ float WT[N * KP];

  // cooperative transpose-stage of W into LDS
  for (int idx = threadIdx.x; idx < K * N; idx += blockDim.x) {
    int k = idx / N;
    int n = idx - k * N;
    WT[n * KP + k] = W[idx];
  }
  __syncthreads();

  int wave   = blockIdx.x * (blockDim.x >> 5) + (threadIdx.x >> 5);
  int lane   = threadIdx.x & 31;
  int tile_m = wave / TILES_N;
  int tile_n = wave - tile_m * TILES_N;
  if (tile_m * 16 >= N_NODES) return;          // wave-uniform

  int half = lane >> 4;                        // 0: lanes 0-15, 1: lanes 16-31
  int l15  = lane & 15;
  int arow = tile_m * 16 + l15;                // A row (M)
  int col  = tile_n * 16 + l15;                // B/D column (N)
  bool colv = FULL ? true : (col < N);
  int  colc = FULL ? col : (colv ? col : (N - 1));  // clamped addr, cndmask only

  const float* __restrict__ xrow = X + arow * K;
  const float* __restrict__ wcol = &WT[colc * KP];
  v8f acc = {};
#pragma unroll
  for (int k0 = 0; k0 < K; k0 += 4) {
    int ka = k0 + 2 * half;
    v2f a = *(const v2f*)(xrow + ka);          // contiguous, 8B aligned
    v2f b = *(const v2f*)(wcol + ka);          // contiguous LDS pair
    if constexpr (!FULL) {
      b.x = colv ? b.x : 0.0f;
      b.y = colv ? b.y : 0.0f;
    }
    acc = __builtin_amdgcn_wmma_f32_16x16x4_f32(false, a, false, b, (short)0, acc,
                                                false, false);
  }
  if (colv) {
    int row0 = tile_m * 16 + half * 8;
#pragma unroll
    for (int r = 0; r < 8; ++r)
      H[(row0 + r) * N + col] = acc[r];
  }
}

// ---------------------------------------------------------------------------
// edge aggregation: AGG[dst] += H[src] * dinv[src]*dinv[dst]
// one wave per edge, 2 floats per lane (F <= 64); native f32 global atomics
// ---------------------------------------------------------------------------
template <int F>
__global__ void gcn_edge_agg(const int* __restrict__ src, const int* __restrict__ dst,
                             const float* __restrict__ dinv, const float* __restrict__ H,
                             float* __restrict__ AGG, int nedges) {
  int tid  = blockIdx.x * blockDim.x + threadIdx.x;
  int e    = tid >> 5;
  if (e >= nedges) return;
  int lane = tid & 31;
  int s = src[e];
  int d = dst[e];
  float norm = dinv[s] * dinv[d];
  int f = lane * 2;
  if (f < F) {
    const float2 hv = *(const float2*)(H + s * F + f);
    float* outp = AGG + d * F + f;
    unsafeAtomicAdd(outp,     hv.x * norm);
    unsafeAtomicAdd(outp + 1, hv.y * norm);
  }
}

// ---------------------------------------------------------------------------
// epilogue: OUT = AGG + H*dinv^2 (self-loop) + bias, optional ReLU (in place)
// ---------------------------------------------------------------------------
template <int F>
__global__ void gcn_epilogue(float* __restrict__ AGG, const float* __restrict__ H,
                             const float* __restrict__ dinv, const float* __restrict__ bias,
                             int do_relu) {
  int i = blockIdx.x * blockDim.x + threadIdx.x;
  if (i >= N_NODES * F) return;
  int node = i / F;                            // F is a power of two -> shift
  int f    = i - node * F;
  float di = dinv[node];
  float v  = AGG[i] + H[i] * di * di + bias[f];
  AGG[i] = do_relu ? fmaxf(v, 0.0f) : v;
}

// ---------------------------------------------------------------------------
// global mean pool accumulation
// ---------------------------------------------------------------------------
__global__ void gcn_pool(const float* __restrict__ H, const int* __restrict__ batch,
                         float* __restrict__ sums, float* __restrict__ cnts, int n) {
  int i = blockIdx.x * blockDim.x + threadIdx.x;
  if (i >= n) return;
  int g = batch[i];
#pragma unroll
  for (int c = 0; c < N_CLS; ++c)
    unsafeAtomicAdd(&sums[g * N_CLS + c], H[i * N_CLS + c]);
  unsafeAtomicAdd(&cnts[g], 1.0f);
}

// ---------------------------------------------------------------------------
// mean + log_softmax per graph
// ---------------------------------------------------------------------------
__global__ void gcn_final(const float* __restrict__ sums, const float* __restrict__ cnts,
                          float* __restrict__ out) {
  int g = blockIdx.x * blockDim.x + threadIdx.x;
  if (g >= N_GRAPHS) return;
  float inv = 1.0f / fmaxf(cnts[g], 1.0f);
  float v[N_CLS];
  float mx = -INFINITY;
#pragma unroll
  for (int c = 0; c < N_CLS; ++c) {
    v[c] = sums[g * N_CLS + c] * inv;
    mx = fmaxf(mx, v[c]);
  }
  float s = 0.0f;
#pragma unroll
  for (int c = 0; c < N_CLS; ++c) s += expf(v[c] - mx);
  float lse = mx + logf(s);
#pragma unroll
  for (int c = 0; c < N_CLS; ++c) out[g * N_CLS + c] = v[c] - lse;
}

// ---------------------------------------------------------------------------
// driver
// ---------------------------------------------------------------------------
extern "C" void kernel_launch(void* const* d_in, const int* in_sizes, int n_in,
                              void* d_out, int out_size, void* d_ws, size_t ws_size,
                              hipStream_t stream) {
  (void)in_sizes; (void)n_in; (void)out_size; (void)ws_size;

  const float* x     = (const float*)d_in[0];
  const int*   eidx  = (const int*)d_in[1];   // edge_index [2, E] flat (harness: integer -> int*)
  const int*   batch = (const int*)d_in[2];
  const float* Wm[5] = {(const float*)d_in[3], (const float*)d_in[5], (const float*)d_in[7],
                        (const float*)d_in[9], (const float*)d_in[11]};
  const float* Bm[5] = {(const float*)d_in[4], (const float*)d_in[6], (const float*)d_in[8],
                        (const float*)d_in[10], (const float*)d_in[12]};
  const int* src = eidx;
  const int* dst = eidx + N_EDGES;

  // workspace layout (~26 MB of floats)
  float* ws   = (float*)d_ws;
  float* dinv = ws;                               // N_NODES
  float* bufA = ws + 50048;                       // N_NODES*HID (gemm output)
  float* bufB = bufA + (size_t)N_NODES * HID;     // N_NODES*HID (agg / layer output)
  float* sums = bufB + (size_t)N_NODES * HID;     // N_GRAPHS*N_CLS
  float* cnts = sums + N_GRAPHS * N_CLS;          // N_GRAPHS (contiguous with sums)

  const int TB = 256;                             // 8 waves / block
  auto blocks = [](long long n, int tb) { return (unsigned)((n + tb - 1) / tb); };

  // --- degree -> dinv ---
  gcn_fill<<<blocks(N_NODES, TB), TB, 0, stream>>>(dinv, 1.0f, N_NODES);
  gcn_deg_accum<<<blocks(N_EDGES, TB), TB, 0, stream>>>(dst, dinv, N_EDGES);
  gcn_dinv_inplace<<<blocks(N_NODES, TB), TB, 0, stream>>>(dinv, N_NODES);

  const int tiles_m = N_NODES / 16;               // 3125 exactly

  // per-layer tail: zero AGG, scatter-add edges, epilogue (self-loop+bias+relu)
  auto run_tail = [&](const float* bias, int N, int relu) {
    gcn_fill<<<blocks((long long)N_NODES * N, TB), TB, 0, stream>>>(bufB, 0.0f, N_NODES * N);
    if (N == HID) {
      gcn_edge_agg<HID><<<blocks((long long)N_EDGES * 32, TB), TB, 0, stream>>>(
          src, dst, dinv, bufA, bufB, N_EDGES);
      gcn_epilogue<HID><<<blocks((long long)N_NODES * N, TB), TB, 0, stream>>>(
          bufB, bufA, dinv, bias, relu);
    } else {
      gcn_edge_agg<N_CLS><<<blocks((long long)N_EDGES * 32, TB), TB, 0, stream>>>(
          src, dst, dinv, bufA, bufB, N_EDGES);
      gcn_epilogue<N_CLS><<<blocks((long long)N_NODES * N, TB), TB, 0, stream>>>(
          bufB, bufA, dinv, bias, relu);
    }
  };

  {  // layer 1: K=128, N=64
    long long thr = (long long)tiles_m * 4 * 32;
    gcn_gemm_wmma<F_IN, HID><<<blocks(thr, TB), TB, 0, stream>>>(x, Wm[0], bufA);
    run_tail(Bm[0], HID, 1);
  }
  for (int l = 1; l <= 3; ++l) {  // layers 2-4: K=64, N=64
    long long thr = (long long)tiles_m * 4 * 32;
    gcn_gemm_wmma<HID, HID><<<blocks(thr, TB), TB, 0, stream>>>(bufB, Wm[l], bufA);
    run_tail(Bm[l], HID, 1);
  }
  {  // layer 5: K=64, N=8
    long long thr = (long long)tiles_m * 1 * 32;
    gcn_gemm_wmma<HID, N_CLS><<<blocks(thr, TB), TB, 0, stream>>>(bufB, Wm[4], bufA);
    run_tail(Bm[4], N_CLS, 0);
  }

  // --- pool + log_softmax ---
  gcn_fill<<<blocks(N_GRAPHS * N_CLS + N_GRAPHS, TB), TB, 0, stream>>>(
      sums, 0.0f, N_GRAPHS * N_CLS + N_GRAPHS);
  gcn_pool<<<blocks(N_NODES, TB), TB, 0, stream>>>(bufB, batch, sums, cnts, N_NODES);
  gcn_final<<<1, 64, 0, stream>>>(sums, cnts, (float*)d_out);
}